// FilterModel_76484777607379
// MI455X (gfx1250) — compile-verified
//
#include <hip/hip_runtime.h>

typedef unsigned int v4u __attribute__((ext_vector_type(4)));
typedef int v8i __attribute__((ext_vector_type(8)));
typedef int v4i __attribute__((ext_vector_type(4)));

#define TILE 64
#define QH 70   // q tile (halo 3)
#define MH 68   // median tile (halo 2)
#define YH 66   // blurred tile (halo 1)
#define NTHREADS 256

__device__ __forceinline__ int reflect101(int v) {
  v = (v < 0) ? -v : v;              // reflect-101 around 0
  return (v > 511) ? (1022 - v) : v; // reflect-101 around 511
}
__device__ __forceinline__ int clamp511(int v) {
  return (v < 0) ? 0 : ((v > 511) ? 511 : v);
}
#define MSORT(a, b) { float _t = fminf(a, b); b = fmaxf(a, b); a = _t; }

__global__ __launch_bounds__(NTHREADS)
void fused_filter_kernel(const float* __restrict__ x, float* __restrict__ out) {
  // LDS pool: [0,4900) = raw-x staging, later reused for y (66*66<=4900)
  //           [4900,9800) = q tile 70x70 ; [9800,14424) = median tile 68x68
  __shared__ float smem[4900 + QH * QH + MH * MH];
  float* A  = smem;            // raw x clipped region, pitch = cw
  float* Dy = smem;            // reused for Gaussian output (stage 3+)
  float* Bq = smem + 4900;     // quantized q, fixed 70x70
  float* Cm = smem + 4900 + QH * QH; // median/255, fixed 68x68

  const int tid = (int)threadIdx.x;
  const int tx0 = (int)blockIdx.x * TILE;
  const int ty0 = (int)blockIdx.y * TILE;
  const float* xp = x + (size_t)blockIdx.z * (512u * 512u);
  float*       op = out + (size_t)blockIdx.z * (512u * 512u);

  // Clipped global rect covering the halo-3 footprint [t0-3, t0+66]
  const int ax0 = (tx0 - 3 < 0) ? 0 : tx0 - 3;
  const int ay0 = (ty0 - 3 < 0) ? 0 : ty0 - 3;
  const int ax1 = (tx0 + 66 > 511) ? 511 : tx0 + 66;
  const int ay1 = (ty0 + 66 > 511) ? 511 : ty0 + 66;
  const int cw = ax1 - ax0 + 1;  // <= 70
  const int ch = ay1 - ay0 + 1;  // <= 70

#if defined(__gfx1250__) && __has_builtin(__builtin_amdgcn_tensor_load_to_lds) && __has_builtin(__builtin_amdgcn_s_wait_tensorcnt)
  // One TDM DMA per workgroup (wave 0 only; scalar branch via readfirstlane).
  if (__builtin_amdgcn_readfirstlane(tid >> 5) == 0) {
    unsigned lds_addr = (unsigned)(unsigned long long)(void*)A;
    unsigned long long ga =
        (unsigned long long)(const void*)(xp + (size_t)ay0 * 512 + ax0);
    v4u g0;
    g0.x = 1u;                                   // count=1, user descriptor
    g0.y = lds_addr;                             // lds_addr [63:32]
    g0.z = (unsigned)ga;                         // global_addr [95:64]
    g0.w = (unsigned)((ga >> 32) & 0x01FFFFFFull) | 0x80000000u; // addr hi + type=2
    v8i g1;
    g1[0] = (int)(2u << 16);                     // data_size = 4 bytes
    g1[1] = (int)((unsigned)cw << 16);           // tensor_dim0 low16
    g1[2] = (int)((unsigned)ch << 16);           // tensor_dim1 low16
    g1[3] = (int)((unsigned)cw << 16);           // tile_dim0
    g1[4] = (int)(unsigned)ch;                   // tile_dim1 (tile_dim2=0)
    g1[5] = 512;                                 // tensor_dim0_stride (elems)
    g1[6] = 0;
    g1[7] = 0;
    v4i g2 = {0, 0, 0, 0};                       // 2D tensor: groups 2/3 unused
    v4i g3 = {0, 0, 0, 0};
    v8i g4 = {0, 0, 0, 0, 0, 0, 0, 0};           // extra slot (clang-23 6-arg form)
    __builtin_amdgcn_tensor_load_to_lds(g0, g1, g2, g3, g4, 0);
    __builtin_amdgcn_s_wait_tensorcnt(0);
  }
#else
  for (int i = tid; i < cw * ch; i += NTHREADS) {
    int r = i / cw;
    A[i] = xp[(size_t)(ay0 + r) * 512 + (size_t)(ax0 + i - r * cw)];
  }
#endif
  __syncthreads();

  // Stage 1: q = floor(clip(x,0,1)*255), halo filled with edge-clamped coords.
  for (int i = tid; i < QH * QH; i += NTHREADS) {
    int ly = i / QH;
    int lx = i - ly * QH;
    int cy = clamp511(ty0 - 3 + ly);
    int cx = clamp511(tx0 - 3 + lx);
    float v = A[(cy - ay0) * cw + (cx - ax0)];
    v = fminf(fmaxf(v, 0.0f), 1.0f);
    Bq[i] = floorf(v * 255.0f);
  }
  __syncthreads();

  // Stage 2: 3x3 median on q (edge padding), /255. Halo via reflect-101 of the
  // *median field* coordinate (matches full-image semantics of the next stage).
  for (int i = tid; i < MH * MH; i += NTHREADS) {
    int ly = i / MH;
    int lx = i - ly * MH;
    int ry = reflect101(ty0 - 2 + ly);
    int rx = reflect101(tx0 - 2 + lx);
    int r0 = clamp511(ry - 1) - (ty0 - 3);
    int r1 = ry - (ty0 - 3);
    int r2 = clamp511(ry + 1) - (ty0 - 3);
    int c0 = clamp511(rx - 1) - (tx0 - 3);
    int c1 = rx - (tx0 - 3);
    int c2 = clamp511(rx + 1) - (tx0 - 3);
    float p0 = Bq[r0 * QH + c0], p1 = Bq[r0 * QH + c1], p2 = Bq[r0 * QH + c2];
    float p3 = Bq[r1 * QH + c0], p4 = Bq[r1 * QH + c1], p5 = Bq[r1 * QH + c2];
    float p6 = Bq[r2 * QH + c0], p7 = Bq[r2 * QH + c1], p8 = Bq[r2 * QH + c2];
    // Paeth 19-comparator median-of-9 network
    MSORT(p1, p2); MSORT(p4, p5); MSORT(p7, p8);
    MSORT(p0, p1); MSORT(p3, p4); MSORT(p6, p7);
    MSORT(p1, p2); MSORT(p4, p5); MSORT(p7, p8);
    MSORT(p0, p3); MSORT(p5, p8); MSORT(p4, p7);
    MSORT(p3, p6); MSORT(p1, p4); MSORT(p2, p5);
    MSORT(p4, p7); MSORT(p4, p2); MSORT(p6, p4);
    MSORT(p4, p2);
    Cm[i] = p4 * (1.0f / 255.0f);
  }
  __syncthreads();

  // Stage 3: torchvision GaussianBlur(3, sigma=1), reflect-101 padding.
  const float G0 = 0.45186276f, G1 = 0.27406862f;
  for (int i = tid; i < YH * YH; i += NTHREADS) {
    int ly = i / YH;
    int lx = i - ly * YH;
    int ry = reflect101(ty0 - 1 + ly);
    int rx = reflect101(tx0 - 1 + lx);
    int r0 = reflect101(ry - 1) - (ty0 - 2);
    int r1 = ry - (ty0 - 2);
    int r2 = reflect101(ry + 1) - (ty0 - 2);
    int c0 = reflect101(rx - 1) - (tx0 - 2);
    int c1 = rx - (tx0 - 2);
    int c2 = reflect101(rx + 1) - (tx0 - 2);
    float h0 = G1 * Cm[r0 * MH + c0] + G0 * Cm[r0 * MH + c1] + G1 * Cm[r0 * MH + c2];
    float h1 = G1 * Cm[r1 * MH + c0] + G0 * Cm[r1 * MH + c1] + G1 * Cm[r1 * MH + c2];
    float h2 = G1 * Cm[r2 * MH + c0] + G0 * Cm[r2 * MH + c1] + G1 * Cm[r2 * MH + c2];
    Dy[i] = G1 * h0 + G0 * h1 + G1 * h2;   // reuses the stage-1 staging buffer
  }
  __syncthreads();

  // Stage 4: unsharp mask: clip(4*y - 3*blur(y), 0, 1)  (== /255-scaled form)
  const int orow = tid >> 4;          // 0..15
  const int ocol = (tid & 15) << 2;   // 0,4,...,60
  for (int k = 0; k < 4; ++k) {
    int ly = orow + 16 * k;           // 0..63
    int gy = ty0 + ly;
    int r0 = reflect101(gy - 1) - (ty0 - 1);
    int r1 = ly + 1;
    int r2 = reflect101(gy + 1) - (ty0 - 1);
    float4 res;
    float* resp = &res.x;
#pragma unroll
    for (int j = 0; j < 4; ++j) {
      int lx = ocol + j;
      int gx = tx0 + lx;
      int c0 = reflect101(gx - 1) - (tx0 - 1);
      int c1 = lx + 1;
      int c2 = reflect101(gx + 1) - (tx0 - 1);
      float h0 = 0.25f * Dy[r0 * YH + c0] + 0.5f * Dy[r0 * YH + c1] + 0.25f * Dy[r0 * YH + c2];
      float h1 = 0.25f * Dy[r1 * YH + c0] + 0.5f * Dy[r1 * YH + c1] + 0.25f * Dy[r1 * YH + c2];
      float h2 = 0.25f * Dy[r2 * YH + c0] + 0.5f * Dy[r2 * YH + c1] + 0.25f * Dy[r2 * YH + c2];
      float blur = 0.25f * h0 + 0.5f * h1 + 0.25f * h2;
      float c = Dy[r1 * YH + c1];
      resp[j] = fminf(fmaxf(4.0f * c - 3.0f * blur, 0.0f), 1.0f);
    }
    *(float4*)(op + (size_t)gy * 512 + (size_t)(tx0 + ocol)) = res;  // b128 store
  }
}

extern "C" void kernel_launch(void* const* d_in, const int* in_sizes, int n_in,
                              void* d_out, int out_size, void* d_ws, size_t ws_size,
                              hipStream_t stream) {
  (void)in_sizes; (void)n_in; (void)out_size; (void)d_ws; (void)ws_size;
  const float* x = (const float*)d_in[0];
  float* out = (float*)d_out;
  dim3 grid(512 / TILE, 512 / TILE, 16 * 3);  // 8 x 8 tiles x 48 planes
  fused_filter_kernel<<<grid, NTHREADS, 0, stream>>>(x, out);
}